// GNN_33406255628772
// MI455X (gfx1250) — compile-verified
//
#include <hip/hip_runtime.h>
#include <hip/hip_bf16.h>

typedef __attribute__((ext_vector_type(2))) float v2f;
typedef __attribute__((ext_vector_type(8))) float v8f;

#define NNODES 50000
#define NEDGES 1600000
#define HCH    64
#define NGRAPH 64
#define COUT   51
#define RTILES 5          // 3125 row tiles = 625 blocks * 5 tiles/wave

// ---------------- Layer 1 (C_IN = 1) ----------------
__global__ void layer1_scatter(const float* __restrict__ x,
                               const int* __restrict__ src,
                               const int* __restrict__ dst,
                               const float* __restrict__ ew,
                               float* __restrict__ agg1, int nE) {
    int e = blockIdx.x * 256 + threadIdx.x;
    if (e < nE) {
        atomicAdd(&agg1[dst[e]], x[src[e]] * ew[e]);
    }
}

__global__ void layer1_dense(const float* __restrict__ x,
                             const float* __restrict__ agg1,
                             const float* __restrict__ w1r,
                             const float* __restrict__ b1,
                             const float* __restrict__ w1n,
                             float* __restrict__ h1, int nN) {
    int idx = blockIdx.x * 256 + threadIdx.x;
    int n = idx >> 6, j = idx & 63;
    if (n < nN) {
        float v = agg1[n] * w1r[j] + b1[j] + x[n] * w1n[j];
        h1[idx] = fmaxf(v, 0.0f);
    }
}

// ---------------- Edge scatter for H=64 layers ----------------
// agg[dst[e]][c] += h[src[e]][c] * ew[e]; 64 lanes per edge -> coalesced
// 256B L2-resident f32 atomic bursts.
__global__ void edge_scatter64(const float* __restrict__ h,
                               const int* __restrict__ src,
                               const int* __restrict__ dst,
                               const float* __restrict__ ew,
                               float* __restrict__ agg, int nE) {
    int e = blockIdx.x * 4 + (threadIdx.x >> 6);
    int c = threadIdx.x & 63;
    if (e < nE) {
        int s = src[e], d = dst[e];
        float w = ew[e];
        atomicAdd(&agg[d * HCH + c], h[s * HCH + c] * w);
    }
}

// ---------------- Fused dual-GEMM via V_WMMA_F32_16X16X4_F32 ----------------
// Each wave owns one 16-column tile; preloads BOTH weight matrices' B
// fragments into 64 VGPRs, then streams RTILES row tiles through them:
// hout = act( agg @ wr + hin @ wn + bias ).
__global__ __launch_bounds__(128)
void gconv_gemm_wmma(const float* __restrict__ agg,   // [N,64]
                     const float* __restrict__ hin,   // [N,64]
                     const float* __restrict__ wr,    // [64,64] row-major [k][j]
                     const float* __restrict__ wn,    // [64,64]
                     const float* __restrict__ bias,  // [64]
                     float* __restrict__ hout,        // [N,64]
                     int do_relu) {
    const int lane    = threadIdx.x & 31;
    const int wave    = threadIdx.x >> 5;       // 0..3 -> column tile
    const int colBase = wave * 16;
    const int r       = lane & 15;              // 0..15
    const int half    = lane >> 4;              // 0 or 1
    const int tile0   = blockIdx.x * RTILES;

    // B (4x16) fragment layout mirrors A: lane half 0 -> K=k,k+1 ; half 1 -> K=k+2,k+3
    v2f bR[16], bN[16];
    #pragma unroll
    for (int kk = 0; kk < 16; ++kk) {
        const int k = kk * 4 + 2 * half;
        bR[kk].x = wr[(k + 0) * HCH + colBase + r];
        bR[kk].y = wr[(k + 1) * HCH + colBase + r];
        bN[kk].x = wn[(k + 0) * HCH + colBase + r];
        bN[kk].y = wn[(k + 1) * HCH + colBase + r];
    }
    const float bj = bias[colBase + r];

    for (int t = 0; t < RTILES; ++t) {
        const int rowBase = (tile0 + t) * 16;
        const int row     = rowBase + r;
        const float* aggRow = agg + (size_t)row * HCH;
        const float* hinRow = hin + (size_t)row * HCH;

        if (t + 1 < RTILES) {
            // prefetch next row tile's A stripes (global_prefetch_b8)
            __builtin_prefetch(aggRow + 16 * HCH, 0, 3);
            __builtin_prefetch(hinRow + 16 * HCH, 0, 3);
        }

        v8f acc = {};
        #pragma unroll
        for (int kk = 0; kk < 16; ++kk) {
            v2f a = *(const v2f*)(aggRow + kk * 4 + 2 * half);
            acc = __builtin_amdgcn_wmma_f32_16x16x4_f32(
                false, a, false, bR[kk], (short)0, acc, false, false);
        }
        #pragma unroll
        for (int kk = 0; kk < 16; ++kk) {
            v2f a = *(const v2f*)(hinRow + kk * 4 + 2 * half);
            acc = __builtin_amdgcn_wmma_f32_16x16x4_f32(
                false, a, false, bN[kk], (short)0, acc, false, false);
        }

        // C/D layout: VGPR i -> row (i + 8*half), col r
        #pragma unroll
        for (int i = 0; i < 8; ++i) {
            int m = rowBase + i + 8 * half;
            float v = acc[i] + bj;
            if (do_relu) v = fmaxf(v, 0.0f);
            hout[m * HCH + colBase + r] = v;
        }
    }
}

// ---------------- Mean pool + head ----------------
__global__ void pool_scatter(const float* __restrict__ h3,
                             const int* __restrict__ batch,
                             float* __restrict__ sums,   // [G,64]
                             float* __restrict__ cnts,   // [G]
                             int nN) {
    int n = blockIdx.x * 4 + (threadIdx.x >> 6);
    int j = threadIdx.x & 63;
    if (n < nN) {
        int g = batch[n];
        atomicAdd(&sums[g * HCH + j], h3[n * HCH + j]);
        if (j == 0) atomicAdd(&cnts[g], 1.0f);
    }
}

__global__ void head_kernel(const float* __restrict__ sums,
                            const float* __restrict__ cnts,
                            const float* __restrict__ wl,  // [64,51]
                            const float* __restrict__ bl,  // [51]
                            float* __restrict__ out) {     // [G,51]
    int g = blockIdx.x;
    int c = threadIdx.x;
    if (c < COUT) {
        float inv = 1.0f / fmaxf(cnts[g], 1.0f);
        float acc = bl[c];
        #pragma unroll 8
        for (int j = 0; j < HCH; ++j)
            acc = fmaf(sums[g * HCH + j] * inv, wl[j * COUT + c], acc);
        out[g * COUT + c] = acc;
    }
}

extern "C" void kernel_launch(void* const* d_in, const int* in_sizes, int n_in,
                              void* d_out, int out_size, void* d_ws, size_t ws_size,
                              hipStream_t stream) {
    const float* x   = (const float*)d_in[0];
    const int*   ei  = (const int*)d_in[1];   // [2,E]
    const int*   bat = (const int*)d_in[2];
    const float* ew  = (const float*)d_in[3];
    const float* w1r = (const float*)d_in[4];
    const float* b1  = (const float*)d_in[5];
    const float* w1n = (const float*)d_in[6];
    const float* w2r = (const float*)d_in[7];
    const float* b2  = (const float*)d_in[8];
    const float* w2n = (const float*)d_in[9];
    const float* w3r = (const float*)d_in[10];
    const float* b3  = (const float*)d_in[11];
    const float* w3n = (const float*)d_in[12];
    const float* wl  = (const float*)d_in[13];
    const float* bl  = (const float*)d_in[14];
    float* out = (float*)d_out;

    const int* src = ei;
    const int* dst = ei + NEDGES;

    // Workspace layout (floats)
    float* ws   = (float*)d_ws;
    float* hA   = ws;                          // [N,64]  h1 / h3
    float* hB   = hA + (size_t)NNODES * HCH;   // [N,64]  h2
    float* agg  = hB + (size_t)NNODES * HCH;   // [N,64]  per-layer aggregate
    float* agg1 = agg + (size_t)NNODES * HCH;  // [N]     layer-1 aggregate
    float* sums = agg1 + NNODES;               // [G,64]
    float* cnts = sums + NGRAPH * HCH;         // [G]

    const int edgeBlocks   = (NEDGES + 255) / 256;
    const int edge64Blocks = (NEDGES + 3) / 4;
    const int denseBlocks  = (NNODES * HCH + 255) / 256;
    const int gemmBlocks   = NNODES / (16 * RTILES);   // 625, exact
    const int poolBlocks   = (NNODES + 3) / 4;

    // ---- Layer 1 ----
    hipMemsetAsync(agg1, 0, sizeof(float) * NNODES, stream);
    layer1_scatter<<<edgeBlocks, 256, 0, stream>>>(x, src, dst, ew, agg1, NEDGES);
    layer1_dense<<<denseBlocks, 256, 0, stream>>>(x, agg1, w1r, b1, w1n, hA, NNODES);

    // ---- Layer 2 ----
    hipMemsetAsync(agg, 0, sizeof(float) * (size_t)NNODES * HCH, stream);
    edge_scatter64<<<edge64Blocks, 256, 0, stream>>>(hA, src, dst, ew, agg, NEDGES);
    gconv_gemm_wmma<<<gemmBlocks, 128, 0, stream>>>(agg, hA, w2r, w2n, b2, hB, 1);

    // ---- Layer 3 ----
    hipMemsetAsync(agg, 0, sizeof(float) * (size_t)NNODES * HCH, stream);
    edge_scatter64<<<edge64Blocks, 256, 0, stream>>>(hB, src, dst, ew, agg, NEDGES);
    gconv_gemm_wmma<<<gemmBlocks, 128, 0, stream>>>(agg, hB, w3r, w3n, b3, hA, 0);

    // ---- Pool + head ----
    hipMemsetAsync(sums, 0, sizeof(float) * (NGRAPH * HCH + NGRAPH), stream);
    pool_scatter<<<poolBlocks, 256, 0, stream>>>(hA, bat, sums, cnts, NNODES);
    head_kernel<<<NGRAPH, 64, 0, stream>>>(sums, cnts, wl, bl, out);
}